// MPPI_8615704396351
// MI455X (gfx1250) — compile-verified
//
#include <hip/hip_runtime.h>
#include <hip/hip_bf16.h>

#define N_ROLL   262144
#define T_STEPS  64
#define JDIM     128            // CD*T
#define DT_C     0.1f
#define INV_TEMP 10.0f
#define MAXC     0.5f
#define MINC    (-0.5f)

typedef __attribute__((ext_vector_type(2))) float v2f;
typedef __attribute__((ext_vector_type(8))) float v8f;

// d_ws float layout: [0] min-cost bits (uint), [1] sum of weights,
// [2..129] weighted-noise accumulators (128), [256..256+N) per-rollout cost.

__global__ void mppi_init(float* ws) {
    int i = threadIdx.x;
    if (i == 0) ((unsigned int*)ws)[0] = 0x7F800000u;   // +inf
    if (i == 1) ws[1] = 0.0f;
    if (i >= 2 && i < 2 + JDIM) ws[i] = 0.0f;
}

// Pass 1: rollout dynamics, cost, and global min reduction.
__global__ void mppi_rollout(const float* __restrict__ state,
                             const float* __restrict__ ctrl,
                             const float* __restrict__ goal,
                             const float* __restrict__ noise,
                             float* __restrict__ ws) {
    int n = blockIdx.x * blockDim.x + threadIdx.x;
    float x = state[0], y = state[1], th = state[2];
    float ccv = ctrl[0], ccw = ctrl[1];

    const float4* vrow = (const float4*)(noise + (size_t)n * JDIM);
    const float4* wrow = (const float4*)(noise + (size_t)n * JDIM + T_STEPS);
#pragma unroll
    for (int q = 0; q < T_STEPS / 4; ++q) {
        float4 v4 = vrow[q];
        float4 w4 = wrow[q];
        float vv[4] = {v4.x, v4.y, v4.z, v4.w};
        float wv[4] = {w4.x, w4.y, w4.z, w4.w};
#pragma unroll
        for (int k = 0; k < 4; ++k) {
            float v = ccv + vv[k];
            float w = ccw + wv[k];
            x += v * __cosf(th) * DT_C;          // v_cos_f32
            y += v * __sinf(th) * DT_C;          // v_sin_f32
            th += w * DT_C;
        }
    }
    float dx = x - goal[0], dy = y - goal[1];
    float cost = sqrtf(dx * dx + dy * dy);
    ws[256 + n] = cost;

    // wave then block min; float bits are order-preserving for cost >= 0
    float m = cost;
    for (int off = 16; off > 0; off >>= 1)
        m = fminf(m, __shfl_down(m, off, 32));
    __shared__ float smin[8];
    int lane = threadIdx.x & 31, wv = threadIdx.x >> 5;
    if (lane == 0) smin[wv] = m;
    __syncthreads();
    if (wv == 0) {
        int nw = blockDim.x >> 5;
        float mm = (lane < nw) ? smin[lane] : 3.4e38f;
        for (int off = 4; off > 0; off >>= 1)
            mm = fminf(mm, __shfl_down(mm, off, 32));
        if (lane == 0)
            atomicMin((unsigned int*)ws, __float_as_uint(mm));
    }
}

// Pass 2: weights + GEMV acc[128] = noise^T * w via V_WMMA_F32_16X16X4_F32.
// A(16x4): A[i][k] = noise[n0+k][j0+i]  (per ISA: lanes 0-15 hold K=0,1 in
// v0,v1; lanes 16-31 hold K=2,3).  B(4x16): B[k][j] = w[n0+k], mirrored K
// split.  D[i][j] = sum_k w[n0+k]*noise[n0+k][j0+i], constant over j.
__global__ void mppi_weighted_sum(const float* __restrict__ noise,
                                  float* __restrict__ ws) {
    const float* cost = ws + 256;
    float cmin = __uint_as_float(((const unsigned int*)ws)[0]);

    int lane = threadIdx.x & 31;
    int half = lane >> 4;          // 0: K=0,1   1: K=2,3
    int m16  = lane & 15;          // row index within A / D column
    int wvid   = (blockIdx.x * blockDim.x + threadIdx.x) >> 5;
    int nwaves = (gridDim.x * blockDim.x) >> 5;

    v8f acc[8];
#pragma unroll
    for (int t = 0; t < 8; ++t) acc[t] = (v8f)0.0f;
    float sw = 0.0f;

    int nchunks = N_ROLL / 4;
    for (int c = wvid; c < nchunks; c += nwaves) {
        int n0 = c * 4;
        int rb = n0 + 2 * half;
        float w0 = __expf(-INV_TEMP * (cost[rb]     - cmin));   // v_exp_f32
        float w1 = __expf(-INV_TEMP * (cost[rb + 1] - cmin));
        v2f b; b.x = w0; b.y = w1;
        sw += w0 + w1;
        const float* r0 = noise + (size_t)rb * JDIM;
#pragma unroll
        for (int t = 0; t < 8; ++t) {
            int j = t * 16 + m16;
            v2f a; a.x = r0[j]; a.y = r0[j + JDIM];   // rollouts rb, rb+1
            acc[t] = __builtin_amdgcn_wmma_f32_16x16x4_f32(
                false, a, false, b, (short)0, acc[t], false, false);
        }
    }

    __shared__ float lacc[JDIM];
    __shared__ float lsw;
    if (threadIdx.x < JDIM) lacc[threadIdx.x] = 0.0f;
    if (threadIdx.x == 0) lsw = 0.0f;
    __syncthreads();

    // D extract: VGPR r -> M=r (lanes 0-15) / M=r+8 (lanes 16-31)
    if (m16 == 0) {
#pragma unroll
        for (int t = 0; t < 8; ++t) {
            int j0 = t * 16 + half * 8;
#pragma unroll
            for (int r = 0; r < 8; ++r)
                atomicAdd(&lacc[j0 + r], acc[t][r]);     // ds_add_f32
        }
    }
    sw += __shfl_down(sw, 16, 32);                       // combine K halves
    if (lane == 0) atomicAdd(&lsw, sw);
    __syncthreads();

    if (threadIdx.x < JDIM) atomicAdd(&ws[2 + threadIdx.x], lacc[threadIdx.x]);
    if (threadIdx.x == 0)   atomicAdd(&ws[1], lsw);
}

// Pass 3: out[c,t] = clip(cc[c] + acc/sumw)
__global__ void mppi_finalize(const float* __restrict__ ctrl,
                              const float* __restrict__ ws,
                              float* __restrict__ out) {
    int j = threadIdx.x;                 // 128 threads
    float sumw = ws[1];
    float val = ctrl[j >> 6] + ws[2 + j] / sumw;
    out[j] = fminf(MAXC, fmaxf(MINC, val));
}

extern "C" void kernel_launch(void* const* d_in, const int* in_sizes, int n_in,
                              void* d_out, int out_size, void* d_ws, size_t ws_size,
                              hipStream_t stream) {
    const float* state = (const float*)d_in[0];   // (3,)
    const float* ctrl  = (const float*)d_in[1];   // (2,1)
    const float* goal  = (const float*)d_in[2];   // (3,)
    const float* noise = (const float*)d_in[3];   // (N, 2, 64)
    float* out = (float*)d_out;                   // (2, 64)
    float* ws  = (float*)d_ws;

    mppi_init<<<1, 256, 0, stream>>>(ws);
    mppi_rollout<<<N_ROLL / 256, 256, 0, stream>>>(state, ctrl, goal, noise, ws);
    mppi_weighted_sum<<<512, 256, 0, stream>>>(noise, ws);
    mppi_finalize<<<1, JDIM, 0, stream>>>(ctrl, ws, out);
}